// MoEAttentionProjection_15204184227985
// MI455X (gfx1250) — compile-verified
//
#include <hip/hip_runtime.h>

typedef __attribute__((ext_vector_type(2))) float v2f;
typedef __attribute__((ext_vector_type(4))) float v4f;
typedef __attribute__((ext_vector_type(8))) float v8f;

#define BB    4
#define SS    4096
#define INF   1024
#define OUTF  1024
#define EE    8
#define NTOK  (BB * SS)      // 16384
#define SCALE 512.0f
#define T3    64             // tokens per block in epilogue

// ---------------------------------------------------------------------------
// k0: zero the logits accumulator (ws is poisoned with 0xAA by the harness)
// ---------------------------------------------------------------------------
__global__ void k0_init(float* __restrict__ logits) {
    if (threadIdx.x < BB * EE) logits[threadIdx.x] = 0.0f;
}

// ---------------------------------------------------------------------------
// k1: fused GEMM  P[t, j] = x[t,:] . W[j,:]   (t = 0..16383, j = 0..15)
//     W rows 0..7  = lora_A[e,0,:]   -> P cols 0..7  = h[t,e]
//     W rows 8..15 = gate_w[e,:]     -> P cols 8..15 = per-token gate dots,
//                                       reduced over tokens into logits[b,e]
// One wave computes one 16x16 output tile via 256x V_WMMA_F32_16X16X4_F32.
// A-frag (16x4 f32): lane L holds A[L%16, k0] , A[L%16, k0+1], k0 = (L<16?0:2)
// B-frag (4x16 f32): lane L holds B[k0, L%16] , B[k0+1, L%16]  (B = W^T)
// D (16x16 f32):     lane L, vgpr r -> row M = r + (L/16)*8, col N = L%16
// ---------------------------------------------------------------------------
__global__ __launch_bounds__(128) void k1_gemm(const float* __restrict__ x,
                                               const float* __restrict__ loraA,
                                               const float* __restrict__ gateW,
                                               float* __restrict__ h,
                                               float* __restrict__ logits) {
    const int wave = threadIdx.x >> 5;
    const int lane = threadIdx.x & 31;
    const int tile = blockIdx.x * 4 + wave;      // 1024 tiles of 16 tokens
    const int t0   = tile << 4;
    const int m    = lane & 15;                  // row (A) / col (B) index
    const int kh   = (lane >> 4) << 1;           // K sub-offset: 0 or 2

    const float* xrow = x + (size_t)(t0 + m) * INF + kh;
    const float* wrow = (m < EE) ? (loraA + (size_t)m * INF + kh)
                                 : (gateW + (size_t)(m - EE) * INF + kh);

    v8f acc = {};
#pragma unroll 8
    for (int k = 0; k < INF; k += 4) {
        v2f a = *(const v2f*)(xrow + k);         // 8B aligned global_load_b64
        v2f b = *(const v2f*)(wrow + k);
        acc = __builtin_amdgcn_wmma_f32_16x16x4_f32(
            /*neg_a=*/false, a, /*neg_b=*/false, b,
            /*c_mod=*/(short)0, acc, /*reuse_a=*/false, /*reuse_b=*/false);
    }

    const int mbase = t0 + ((lane >> 4) << 3);   // first of this lane's 8 rows
    if (m < EE) {
        // columns 0..7: LoRA dots -> h[token, e]
#pragma unroll
        for (int r = 0; r < 8; ++r)
            h[(size_t)(mbase + r) * EE + m] = acc[r];
    } else {
        // columns 8..15: gate dots -> sum this lane's 8 rows, accumulate
        float p = (acc[0] + acc[1]) + (acc[2] + acc[3]) +
                  (acc[4] + acc[5]) + (acc[6] + acc[7]);
        const int b = t0 >> 12;                  // t0 / 4096 (tiles never straddle)
        atomicAdd(&logits[b * EE + (m - EE)], p);
    }
}

// ---------------------------------------------------------------------------
// k2: per-batch softmax + top-2 -> coef[b,e] = SCALE*score (0 if unselected)
// ---------------------------------------------------------------------------
__global__ void k2_gate(const float* __restrict__ logits,
                        const float* __restrict__ gateB,
                        float* __restrict__ coef) {
    const int lane = threadIdx.x;
    if (lane < BB * EE) coef[lane] = 0.0f;       // wave-ordered before writes below
    if (lane < BB) {
        float l[EE];
        float mx = -1e30f;
        for (int e = 0; e < EE; ++e) {
            l[e] = logits[lane * EE + e] * (1.0f / (float)SS) + gateB[e];
            mx = fmaxf(mx, l[e]);
        }
        float p[EE], sum = 0.0f;
        for (int e = 0; e < EE; ++e) { p[e] = expf(l[e] - mx); sum += p[e]; }
        const float inv = 1.0f / sum;
        // top-2; strict '>' keeps first index on ties (matches jax top_k)
        int i1 = 0;
        for (int e = 1; e < EE; ++e) if (l[e] > l[i1]) i1 = e;
        int i2 = (i1 == 0) ? 1 : 0;
        for (int e = 0; e < EE; ++e) if (e != i1 && l[e] > l[i2]) i2 = e;
        coef[lane * EE + i1] = SCALE * p[i1] * inv;
        coef[lane * EE + i2] = SCALE * p[i2] * inv;
    }
}

// ---------------------------------------------------------------------------
// k3: out[t,:] = sum_e (coef[b,e]*h[t,e]) * lora_B[e,:]
// 256 threads cover 1024 cols (float4 each); lora_B held in registers and
// reused across T3 tokens; h*coef staged in LDS; 512B coalesced stores.
// ---------------------------------------------------------------------------
__global__ __launch_bounds__(256) void k3_out(const float* __restrict__ h,
                                              const float* __restrict__ loraB,
                                              const float* __restrict__ coef,
                                              float* __restrict__ out) {
    __shared__ float wh[T3 * EE];
    const int t0  = blockIdx.x * T3;
    const int b   = t0 / SS;
    const int tid = threadIdx.x;
    const int c0  = tid * 4;

    v4f Breg[EE];
#pragma unroll
    for (int e = 0; e < EE; ++e)
        Breg[e] = *(const v4f*)(loraB + (size_t)e * OUTF + c0);

    for (int i = tid; i < T3 * EE; i += 256) {
        const int e = i & 7;
        wh[i] = h[(size_t)t0 * EE + i] * coef[b * EE + e];
    }
    __syncthreads();

    for (int t = 0; t < T3; ++t) {
        v4f o = {};
#pragma unroll
        for (int e = 0; e < EE; ++e) {
            const float w = wh[t * EE + e];
            o += Breg[e] * w;
        }
        *(v4f*)(out + (size_t)(t0 + t) * OUTF + c0) = o;
    }
}

// ---------------------------------------------------------------------------
extern "C" void kernel_launch(void* const* d_in, const int* in_sizes, int n_in,
                              void* d_out, int out_size, void* d_ws, size_t ws_size,
                              hipStream_t stream) {
    const float* x     = (const float*)d_in[0];   // [B,S,IN]
    const float* loraA = (const float*)d_in[1];   // [E,1,IN]
    const float* loraB = (const float*)d_in[2];   // [E,OUT,1]
    const float* gateW = (const float*)d_in[3];   // [E,IN]
    const float* gateB = (const float*)d_in[4];   // [E]
    float* out = (float*)d_out;                   // [B,S,OUT]

    // workspace layout: h[16384*8] | logits[32] | coef[32]   (~512.3 KB)
    float* h      = (float*)d_ws;
    float* logits = h + (size_t)NTOK * EE;
    float* coef   = logits + BB * EE;

    k0_init<<<1, 64, 0, stream>>>(logits);
    k1_gemm<<<256, 128, 0, stream>>>(x, loraA, gateW, h, logits);
    k2_gate<<<1, 32, 0, stream>>>(logits, gateB, coef);
    k3_out<<<NTOK / T3, 256, 0, stream>>>(h, loraB, coef, out);
}